// NeuralMemory_9079560864297
// MI455X (gfx1250) — compile-verified
//
#include <hip/hip_runtime.h>

// ---------------------------------------------------------------------------
// NeuralMemory (Titans-style) for MI455X / gfx1250.
// fp32 GEMMs via V_WMMA_F32_16X16X4_F32 with LDS double-buffered tiling.
// Dims (fixed): Q=1024, K=512, V=512, H=1024, B=8, L=4096
// ---------------------------------------------------------------------------

typedef float v2f __attribute__((ext_vector_type(2)));
typedef float v8f __attribute__((ext_vector_type(8)));

#define DIM_Q 1024
#define DIM_K 512
#define DIM_V 512
#define DIM_H 1024
#define NB    8
#define NL    4096

#define ALPHA_FORGET 0.01f
#define THETA_LR     0.01f
#define ETA_MOM      0.9f

// Epilogue modes
#define EP_NONE      0
#define EP_BIAS      1
#define EP_BIAS_RELU 2
#define EP_DPRED     3   // (acc + bias - target) * scale
#define EP_RELUMASK  4   // acc * (maskSrc > 0)

#define LDS_STRIDE 20    // 16 K-values + pad: keeps ds_load_b64 aligned & conflict-free

// ---------------------------------------------------------------------------
// Tiled fp32 WMMA GEMM:  C[m,n] = epilogue( sum_k A(m,k)*B(k,n) )
//   TA=false: A(m,k) = A[m*lda + k]   TA=true: A(m,k) = A[k*lda + m]
//   TB=false: B(k,n) = B[k*ldb + n]   TB=true: B(k,n) = B[n*ldb + k]
// Block = 4 wave32 -> 64x64 tile; wave = 2x2 fragments of 16x16; K-chunk 16.
// Requires M%64==0, N%64==0, K%16==0 (all shapes here satisfy this).
// LDS layout: As[buf][m][k], Bs[buf][n][k] -> fragment reads are ds_load_b64.
// ---------------------------------------------------------------------------
template<bool TA, bool TB>
__global__ __launch_bounds__(128)
void wmma_gemm_f32(const float* __restrict__ A, long lda,
                   const float* __restrict__ Bm, long ldb,
                   float* __restrict__ C, int ldc,
                   int M, int N, int K,
                   const float* __restrict__ bias,
                   const float* __restrict__ target,
                   const float* __restrict__ maskSrc,
                   float scale, int mode)
{
    __shared__ __align__(16) float As[2][64][LDS_STRIDE];
    __shared__ __align__(16) float Bs[2][64][LDS_STRIDE];

    const int t    = threadIdx.x;
    const int lane = t & 31;
    const int wave = t >> 5;
    const int half = lane >> 4;   // K-pair select / upper-8-rows select
    const int lrow = lane & 15;   // fragment row (A) / col (B,C)
    const int m0   = blockIdx.y * 64;
    const int n0   = blockIdx.x * 64;
    const int wmL  = (wave >> 1) * 32;
    const int wnL  = (wave & 1) * 32;

    v8f acc[2][2] = {};

    auto loadA = [&](int k0, int buf) {
        if (!TA) {
            // contiguous in k: vector global load + vector LDS store
            const int row = t >> 1, cq = (t & 1) * 8;
            const float4 f0 = *(const float4*)&A[(long)(m0 + row) * lda + (k0 + cq)];
            const float4 f1 = *(const float4*)&A[(long)(m0 + row) * lda + (k0 + cq + 4)];
            if (k0 + 32 < K)  // two stages ahead -> global_prefetch_b8
                __builtin_prefetch(&A[(long)(m0 + row) * lda + (k0 + 32 + cq)], 0, 1);
            *(float4*)&As[buf][row][cq]     = f0;
            *(float4*)&As[buf][row][cq + 4] = f1;
        } else {
            // contiguous in m: vector global load, scatter-transpose into LDS
            const int kk = (t >> 4) * 2, mq = (t & 15) * 4;
#pragma unroll
            for (int u = 0; u < 2; ++u) {
                const float4 f = *(const float4*)&A[(long)(k0 + kk + u) * lda + (m0 + mq)];
                As[buf][mq + 0][kk + u] = f.x;
                As[buf][mq + 1][kk + u] = f.y;
                As[buf][mq + 2][kk + u] = f.z;
                As[buf][mq + 3][kk + u] = f.w;
            }
        }
    };
    auto loadB = [&](int k0, int buf) {
        if (!TB) {
            // contiguous in n: vector global load, scatter-transpose into LDS
            const int kk = (t >> 4) * 2, nq = (t & 15) * 4;
#pragma unroll
            for (int u = 0; u < 2; ++u) {
                const float4 f = *(const float4*)&Bm[(long)(k0 + kk + u) * ldb + (n0 + nq)];
                Bs[buf][nq + 0][kk + u] = f.x;
                Bs[buf][nq + 1][kk + u] = f.y;
                Bs[buf][nq + 2][kk + u] = f.z;
                Bs[buf][nq + 3][kk + u] = f.w;
            }
        } else {
            // contiguous in k: vector global load + vector LDS store
            const int row = t >> 1, cq = (t & 1) * 8;
            const float4 f0 = *(const float4*)&Bm[(long)(n0 + row) * ldb + (k0 + cq)];
            const float4 f1 = *(const float4*)&Bm[(long)(n0 + row) * ldb + (k0 + cq + 4)];
            if (k0 + 32 < K)
                __builtin_prefetch(&Bm[(long)(n0 + row) * ldb + (k0 + 32 + cq)], 0, 1);
            *(float4*)&Bs[buf][row][cq]     = f0;
            *(float4*)&Bs[buf][row][cq + 4] = f1;
        }
    };

    loadA(0, 0);
    loadB(0, 0);
    __syncthreads();

    const int nstages = K / 16;
    for (int st = 0; st < nstages; ++st) {
        const int cur = st & 1;
        if (st + 1 < nstages) {           // fill next buffer while computing
            loadA((st + 1) * 16, cur ^ 1);
            loadB((st + 1) * 16, cur ^ 1);
        }
#pragma unroll
        for (int ks = 0; ks < 16; ks += 4) {
            const int ka = ks + 2 * half; // lanes 0-15: K={ks,ks+1}; 16-31: K={ks+2,ks+3}
            v2f a[2], b[2];
#pragma unroll
            for (int r = 0; r < 2; ++r)
                a[r] = *(const v2f*)&As[cur][wmL + r * 16 + lrow][ka];
#pragma unroll
            for (int c = 0; c < 2; ++c)
                b[c] = *(const v2f*)&Bs[cur][wnL + c * 16 + lrow][ka];
#pragma unroll
            for (int r = 0; r < 2; ++r)
#pragma unroll
                for (int c = 0; c < 2; ++c)
                    acc[r][c] = __builtin_amdgcn_wmma_f32_16x16x4_f32(
                        /*neg_a=*/false, a[r], /*neg_b=*/false, b[c],
                        /*c_mod=*/(short)0, acc[r][c],
                        /*reuse_a=*/false, /*reuse_b=*/false);
        }
        __syncthreads();
    }

    // Epilogue + store. C/D layout: vgpr j holds row (8*half + j), col = lrow.
#pragma unroll
    for (int r = 0; r < 2; ++r) {
#pragma unroll
        for (int c = 0; c < 2; ++c) {
#pragma unroll
            for (int j = 0; j < 8; ++j) {
                const int row = m0 + wmL + r * 16 + half * 8 + j;
                const int col = n0 + wnL + c * 16 + lrow;
                const long idx = (long)row * ldc + col;
                float v = acc[r][c][j];
                if (mode == EP_BIAS) {
                    v += bias[col];
                } else if (mode == EP_BIAS_RELU) {
                    v += bias[col];
                    v = v > 0.0f ? v : 0.0f;
                } else if (mode == EP_DPRED) {
                    v = (v + bias[col] - target[idx]) * scale;
                } else if (mode == EP_RELUMASK) {
                    v = (maskSrc[idx] > 0.0f) ? v : 0.0f;
                }
                C[idx] = v;
            }
        }
    }
}

// ---------------------------------------------------------------------------
// Deterministic split-K column sum (bias gradients): part[ch][c] then finalize.
// ---------------------------------------------------------------------------
#define CS_CHUNKS 16
__global__ void colsum_part_kernel(const float* __restrict__ X, float* __restrict__ part,
                                   int rows, int cols)
{
    const int c = blockIdx.x * blockDim.x + threadIdx.x;
    const int ch = blockIdx.y;
    if (c >= cols) return;
    const int span = rows / CS_CHUNKS;
    const int r0 = ch * span;
    float s = 0.0f;
    for (int r = r0; r < r0 + span; ++r) s += X[(long)r * cols + c];
    part[(long)ch * cols + c] = s;
}

__global__ void colsum_fin_kernel(const float* __restrict__ part, float* __restrict__ out,
                                  int cols)
{
    const int c = blockIdx.x * blockDim.x + threadIdx.x;
    if (c >= cols) return;
    float s = 0.0f;
    for (int ch = 0; ch < CS_CHUNKS; ++ch) s += part[(long)ch * cols + c];
    out[c] = s;
}

// Gated momentum + weight-decay update (reference mask semantics):
//   new_S = eta*S - theta*g ; new_p = (1-alpha)*p + new_S ; commit iff gate>0.5
__global__ void update_param_kernel(float* __restrict__ p, float* __restrict__ S,
                                    const float* __restrict__ g, int n,
                                    const float* __restrict__ gate, int step)
{
    const int i = blockIdx.x * blockDim.x + threadIdx.x;
    if (i >= n) return;
    const bool m = gate[step] > 0.5f;
    const float s_new = ETA_MOM * S[i] - THETA_LR * g[i];
    const float p_new = (1.0f - ALPHA_FORGET) * p[i] + s_new;
    if (m) { S[i] = s_new; p[i] = p_new; }
}

__global__ void copy_kernel(const float* __restrict__ src, float* __restrict__ dst, long n)
{
    const long i = (long)blockIdx.x * blockDim.x + threadIdx.x;
    if (i < n) dst[i] = src[i];
}

// ---------------------------------------------------------------------------
// Host-side orchestration
// ---------------------------------------------------------------------------
static inline void gemm(hipStream_t s, bool ta, bool tb,
                        const float* A, long lda,
                        const float* B, long ldb,
                        float* C, int ldc, int M, int N, int K,
                        const float* bias, const float* target, const float* mask,
                        float scale, int mode)
{
    dim3 grid(N / 64, M / 64), block(128);
    if (!ta && !tb)
        wmma_gemm_f32<false, false><<<grid, block, 0, s>>>(A, lda, B, ldb, C, ldc, M, N, K,
                                                           bias, target, mask, scale, mode);
    else if (ta && !tb)
        wmma_gemm_f32<true, false><<<grid, block, 0, s>>>(A, lda, B, ldb, C, ldc, M, N, K,
                                                          bias, target, mask, scale, mode);
    else
        wmma_gemm_f32<false, true><<<grid, block, 0, s>>>(A, lda, B, ldb, C, ldc, M, N, K,
                                                          bias, target, mask, scale, mode);
}

static inline void colsum(hipStream_t s, const float* X, float* part, float* out,
                          int rows, int cols)
{
    dim3 g1((cols + 255) / 256, CS_CHUNKS);
    colsum_part_kernel<<<g1, 256, 0, s>>>(X, part, rows, cols);
    colsum_fin_kernel<<<(cols + 255) / 256, 256, 0, s>>>(part, out, cols);
}

static inline void dcopy(hipStream_t s, const float* src, float* dst, long n)
{
    copy_kernel<<<(unsigned)((n + 255) / 256), 256, 0, s>>>(src, dst, n);
}

extern "C" void kernel_launch(void* const* d_in, const int* in_sizes, int n_in,
                              void* d_out, int out_size, void* d_ws, size_t ws_size,
                              hipStream_t stream)
{
    (void)in_sizes; (void)n_in; (void)out_size; (void)ws_size;

    const float* query   = (const float*)d_in[0];   // (B,L,K)
    const float* content = (const float*)d_in[1];   // (B,L,Q)
    const float* gate    = (const float*)d_in[2];   // (B,)
    const float* W1 = (const float*)d_in[3];        // (K,H)
    const float* b1 = (const float*)d_in[4];        // (H,)
    const float* W2 = (const float*)d_in[5];        // (H,V)
    const float* b2 = (const float*)d_in[6];        // (V,)
    const float* Wk = (const float*)d_in[7];        // (Q,K)
    const float* bk = (const float*)d_in[8];        // (K,)
    const float* Wv = (const float*)d_in[9];        // (Q,V)
    const float* bv = (const float*)d_in[10];       // (V,)
    const float* S_W1 = (const float*)d_in[11];
    const float* S_b1 = (const float*)d_in[12];
    const float* S_W2 = (const float*)d_in[13];
    const float* S_b2 = (const float*)d_in[14];

    float* out = (float*)d_out;                     // retrieved first, then params/S

    // Workspace layout (floats). Param region matches output order for one copy.
    float* ws   = (float*)d_ws;
    float* pW1  = ws;
    float* pb1  = pW1 + (long)DIM_K * DIM_H;
    float* pW2  = pb1 + DIM_H;
    float* pb2  = pW2 + (long)DIM_H * DIM_V;
    float* sW1  = pb2 + DIM_V;
    float* sb1  = sW1 + (long)DIM_K * DIM_H;
    float* sW2  = sb1 + DIM_H;
    float* sb2  = sW2 + (long)DIM_H * DIM_V;
    const long PARAM_FLOATS = 2L * ((long)DIM_K * DIM_H + DIM_H + (long)DIM_H * DIM_V + DIM_V);

    float* h     = sb2 + DIM_V;                     // (L,H)
    float* dh    = h     + (long)NL * DIM_H;        // (L,H)
    float* dpred = dh    + (long)NL * DIM_H;        // (L,V)
    float* kbuf  = dpred + (long)NL * DIM_V;        // (L,K)
    float* vbuf  = kbuf  + (long)NL * DIM_K;        // (L,V)
    float* gW1   = vbuf  + (long)NL * DIM_V;        // (K,H)
    float* gW2   = gW1   + (long)DIM_K * DIM_H;     // (H,V)
    float* gb1   = gW2   + (long)DIM_H * DIM_V;     // (H,)
    float* gb2   = gb1   + DIM_H;                   // (V,)
    float* part  = gb2   + DIM_V;                   // (CS_CHUNKS, max(H,V))

    const float dpred_scale = 2.0f / ((float)NL * (float)DIM_V);

    // ---- init working params / momentum from inputs ----
    dcopy(stream, W1, pW1, (long)DIM_K * DIM_H);
    dcopy(stream, b1, pb1, DIM_H);
    dcopy(stream, W2, pW2, (long)DIM_H * DIM_V);
    dcopy(stream, b2, pb2, DIM_V);
    dcopy(stream, S_W1, sW1, (long)DIM_K * DIM_H);
    dcopy(stream, S_b1, sb1, DIM_H);
    dcopy(stream, S_W2, sW2, (long)DIM_H * DIM_V);
    dcopy(stream, S_b2, sb2, DIM_V);

    // ---- retrieval with pre-update params: out_b = relu(q_b@W1+b1)@W2+b2 ----
    for (int b = 0; b < NB; ++b) {
        const float* qb = query + (long)b * NL * DIM_K;
        float* ob = out + (long)b * NL * DIM_V;
        gemm(stream, false, false, qb, DIM_K, W1, DIM_H, h, DIM_H,
             NL, DIM_H, DIM_K, b1, nullptr, nullptr, 0.0f, EP_BIAS_RELU);
        gemm(stream, false, false, h, DIM_H, W2, DIM_V, ob, DIM_V,
             NL, DIM_V, DIM_H, b2, nullptr, nullptr, 0.0f, EP_BIAS);
    }

    // ---- sequential gated SGD-with-momentum scan over the B samples ----
    for (int i = 0; i < NB; ++i) {
        const float* ci = content + (long)i * NL * DIM_Q;

        // k_i = c_i@Wk + bk ; v_i = c_i@Wv + bv
        gemm(stream, false, false, ci, DIM_Q, Wk, DIM_K, kbuf, DIM_K,
             NL, DIM_K, DIM_Q, bk, nullptr, nullptr, 0.0f, EP_BIAS);
        gemm(stream, false, false, ci, DIM_Q, Wv, DIM_V, vbuf, DIM_V,
             NL, DIM_V, DIM_Q, bv, nullptr, nullptr, 0.0f, EP_BIAS);

        // forward: h = relu(k_i@W1c + b1c)
        gemm(stream, false, false, kbuf, DIM_K, pW1, DIM_H, h, DIM_H,
             NL, DIM_H, DIM_K, pb1, nullptr, nullptr, 0.0f, EP_BIAS_RELU);

        // dpred = 2*(h@W2c + b2c - v_i)/(L*V)
        gemm(stream, false, false, h, DIM_H, pW2, DIM_V, dpred, DIM_V,
             NL, DIM_V, DIM_H, pb2, vbuf, nullptr, dpred_scale, EP_DPRED);

        // gW2 = h^T @ dpred  (transpose-A)
        gemm(stream, true, false, h, DIM_H, dpred, DIM_V, gW2, DIM_V,
             DIM_H, DIM_V, NL, nullptr, nullptr, nullptr, 0.0f, EP_NONE);
        colsum(stream, dpred, part, gb2, NL, DIM_V);

        // dh = (dpred @ W2c^T) * (h > 0)   (transpose-B)
        gemm(stream, false, true, dpred, DIM_V, pW2, DIM_V, dh, DIM_H,
             NL, DIM_H, DIM_V, nullptr, nullptr, h, 0.0f, EP_RELUMASK);

        // gW1 = k_i^T @ dh  (transpose-A)
        gemm(stream, true, false, kbuf, DIM_K, dh, DIM_H, gW1, DIM_H,
             DIM_K, DIM_H, NL, nullptr, nullptr, nullptr, 0.0f, EP_NONE);
        colsum(stream, dh, part, gb1, NL, DIM_H);

        // gated momentum + forgetting updates
        update_param_kernel<<<(DIM_K * DIM_H + 255) / 256, 256, 0, stream>>>(pW1, sW1, gW1, DIM_K * DIM_H, gate, i);
        update_param_kernel<<<(DIM_H + 255) / 256, 256, 0, stream>>>(pb1, sb1, gb1, DIM_H, gate, i);
        update_param_kernel<<<(DIM_H * DIM_V + 255) / 256, 256, 0, stream>>>(pW2, sW2, gW2, DIM_H * DIM_V, gate, i);
        update_param_kernel<<<(DIM_V + 255) / 256, 256, 0, stream>>>(pb2, sb2, gb2, DIM_V, gate, i);
    }

    // ---- write back final params + momentum (contiguous, output order) ----
    dcopy(stream, pW1, out + (long)NB * NL * DIM_V, PARAM_FLOATS);
}